// HungarianMatcher_20736102105833
// MI455X (gfx1250) — compile-verified
//
#include <hip/hip_runtime.h>
#include <hip/hip_bf16.h>
#include <math.h>

// Problem constants (from the reference)
#define NUM_CLASSES 20
#define QPC        81                    // queries per class
#define NQ         (NUM_CLASSES * QPC)   // 1620 queries
#define NCL        21                    // logit classes
#define TPB        20                    // targets per batch

// ---- gfx1250 async global->LDS path (ASYNCcnt), probed via __has_builtin ----
#if defined(__has_builtin)
#  if __has_builtin(__builtin_amdgcn_global_load_async_to_lds_b128) && \
      __has_builtin(__builtin_amdgcn_global_load_async_to_lds_b32)  && \
      __has_builtin(__builtin_amdgcn_s_wait_asynccnt)
#    define HAVE_ASYNC_LDS 1
#  endif
#endif
#ifndef HAVE_ASYNC_LDS
#  define HAVE_ASYNC_LDS 0
#endif

#define AS_GLOBAL __attribute__((address_space(1)))
#define AS_LDS    __attribute__((address_space(3)))
typedef int v4i_gcc __attribute__((vector_size(16)));   // matches builtin param

// ---------------------------------------------------------------------------
// Kernel A: per-query class cost  cc[b*NQ + q] = -softmax(logits[b,q,:])[NCL-1]
// One block = 256 queries. The block's 256*21 floats are staged into LDS —
// on gfx1250 via GLOBAL_LOAD_ASYNC_TO_LDS_B128 (no VGPR round-trip, drained
// with s_wait_asynccnt) — then each thread reduces its own 21 values.
// LDS reads at stride 21 dwords are bank-conflict free (gcd(21,64) == 1).
// ---------------------------------------------------------------------------
__global__ __launch_bounds__(256) void class_cost_kernel(
    const float* __restrict__ logits, float* __restrict__ ccost, int total_q)
{
    __shared__ __align__(16) float sh[256 * NCL];   // 21504 B
    const int tid   = threadIdx.x;
    const int qbase = blockIdx.x * 256;
    const int nq    = min(256, total_q - qbase);
    const int ne    = nq * NCL;
    const float* __restrict__ src = logits + (size_t)qbase * NCL;

#if HAVE_ASYNC_LDS
    // Generic->AS1 is a bit-identical reinterpret; generic LDS addr truncated
    // to 32 bits is the AS3 offset (flat->local lowering is a trunc).
    const unsigned long long gbase = (unsigned long long)(const void*)src;
    const unsigned int       lbase = (unsigned int)(unsigned long long)(const void*)sh;
    const int ne4 = ne >> 2;                         // 16-byte chunks
    for (int i4 = tid; i4 < ne4; i4 += 256) {
        __builtin_amdgcn_global_load_async_to_lds_b128(
            (AS_GLOBAL v4i_gcc*)(gbase + 16ull * (unsigned long long)i4),
            (AS_LDS    v4i_gcc*)(lbase + 16u  * (unsigned int)i4), 0, 0);
    }
    for (int i = (ne4 << 2) + tid; i < ne; i += 256) {   // dword remainder
        __builtin_amdgcn_global_load_async_to_lds_b32(
            (AS_GLOBAL int*)(gbase + 4ull * (unsigned long long)i),
            (AS_LDS    int*)(lbase + 4u  * (unsigned int)i), 0, 0);
    }
    __builtin_amdgcn_s_wait_asynccnt(0);
    __syncthreads();
#else
    // Fallback: synchronous coalesced staging + gfx1250 prefetch
    __builtin_prefetch(src + (ne >> 1) + tid, 0, 0);
    for (int i = tid; i < ne; i += 256) sh[i] = src[i];
    __syncthreads();
#endif

    if (tid < nq) {
        const float* v = sh + tid * NCL;
        float m = v[0];
#pragma unroll
        for (int i = 1; i < NCL; ++i) m = fmaxf(m, v[i]);
        float s = 0.f;
#pragma unroll
        for (int i = 0; i < NCL; ++i) s += expf(v[i] - m);
        ccost[qbase + tid] = -(expf(v[NCL - 1] - m) / s);
    }
}

// ---------------------------------------------------------------------------
// Kernel B: one wave32 per (batch, target) pair.
// Each lane evaluates up to 3 of the 81 class-local queries, then a
// shfl_xor argmin reduction (wave32!) with first-index tie-breaking.
// ---------------------------------------------------------------------------
__global__ __launch_bounds__(128) void match_kernel(
    const float* __restrict__ boxes,   // (B, NQ, 6) cxcyczwhd
    const float* __restrict__ tboxes,  // (B, TPB, 6)
    const int*   __restrict__ labels,  // (B, TPB)
    const float* __restrict__ ccost,   // (B, NQ)
    int*         __restrict__ out,     // (B, TPB, 2) int32
    int npairs)
{
    const int wid  = (blockIdx.x * 128 + threadIdx.x) >> 5;   // (b,t) pair id
    const int lane = threadIdx.x & 31;
    if (wid >= npairs) return;

    const int b = wid / TPB;
    int tc = labels[wid] - 1;
    tc = tc < 0 ? 0 : (tc > NUM_CLASSES - 1 ? NUM_CLASSES - 1 : tc);

    // Target box -> corners + volume (uniform per wave)
    const float* tb = tboxes + (size_t)wid * 6;
    const float tcx = tb[0], tcy = tb[1], tcz = tb[2];
    const float tsx = tb[3], tsy = tb[4], tsz = tb[5];
    const float tl0 = tcx - 0.5f * tsx, th0 = tcx + 0.5f * tsx;
    const float tl1 = tcy - 0.5f * tsy, th1 = tcy + 0.5f * tsy;
    const float tl2 = tcz - 0.5f * tsz, th2 = tcz + 0.5f * tsz;
    const float vol2 = tsx * tsy * tsz;

    const size_t qbase = (size_t)b * NQ + (size_t)tc * QPC;
    float bestc = INFINITY;
    int   bestq = QPC;

#pragma unroll
    for (int k = 0; k < 3; ++k) {
        const int q = lane + 32 * k;
        if (q < QPC) {
            const float* pb = boxes + (qbase + q) * 6;
            const float cx = pb[0], cy = pb[1], cz = pb[2];
            const float sx = pb[3], sy = pb[4], sz = pb[5];

            // class cost + L1 bbox cost (unclipped boxes, per reference)
            float cost = ccost[qbase + q]
                       + fabsf(cx - tcx) + fabsf(cy - tcy) + fabsf(cz - tcz)
                       + fabsf(sx - tsx) + fabsf(sy - tsy) + fabsf(sz - tsz);

            // GIoU uses clip(pred, 0, inf) on all 6 components
            const float px = fmaxf(cx, 0.f), py = fmaxf(cy, 0.f), pz = fmaxf(cz, 0.f);
            const float qx = fmaxf(sx, 0.f), qy = fmaxf(sy, 0.f), qz = fmaxf(sz, 0.f);
            const float l0 = px - 0.5f * qx, h0 = px + 0.5f * qx;
            const float l1 = py - 0.5f * qy, h1 = py + 0.5f * qy;
            const float l2 = pz - 0.5f * qz, h2 = pz + 0.5f * qz;
            const float vol1 = qx * qy * qz;

            const float i0 = fminf(h0, th0) - fmaxf(l0, tl0);
            const float i1 = fminf(h1, th1) - fmaxf(l1, tl1);
            const float i2 = fminf(h2, th2) - fmaxf(l2, tl2);
            const float inter = fmaxf(i0, 0.f) * fmaxf(i1, 0.f) * fmaxf(i2, 0.f);
            const float uni = vol1 + vol2 - inter;
            const float iou = inter / uni;

            const float e0 = fmaxf(h0, th0) - fminf(l0, tl0);
            const float e1 = fmaxf(h1, th1) - fminf(l1, tl1);
            const float e2 = fmaxf(h2, th2) - fminf(l2, tl2);
            const float vole = fmaxf(e0, 0.f) * fmaxf(e1, 0.f) * fmaxf(e2, 0.f);

            cost -= iou - (vole - uni) / vole;      // cost += -giou

            if (cost < bestc) { bestc = cost; bestq = q; }  // keeps lowest q on ties
        }
    }

    // wave32 argmin reduction; prefer lower index on exact ties
#pragma unroll
    for (int off = 16; off > 0; off >>= 1) {
        const float oc = __shfl_xor(bestc, off, 32);
        const int   oq = __shfl_xor(bestq, off, 32);
        if (oc < bestc || (oc == bestc && oq < bestq)) { bestc = oc; bestq = oq; }
    }

    if (lane == 0) {
        out[wid * 2 + 0] = tc;
        out[wid * 2 + 1] = tc * QPC + bestq;
    }
}

// ---------------------------------------------------------------------------
extern "C" void kernel_launch(void* const* d_in, const int* in_sizes, int n_in,
                              void* d_out, int out_size, void* d_ws, size_t ws_size,
                              hipStream_t stream)
{
    const float* pred_logits = (const float*)d_in[0];  // (B, NQ, 21) f32
    const float* pred_boxes  = (const float*)d_in[1];  // (B, NQ, 6)  f32
    const float* tgt_boxes   = (const float*)d_in[2];  // (B, TPB, 6) f32
    /* d_in[3] = anchors — unused by the reference */
    const int*   tgt_labels  = (const int*)d_in[4];    // (B, TPB)    i32
    int* out = (int*)d_out;                            // (B, TPB, 2) i32

    const int B       = in_sizes[4] / TPB;             // 512
    const int total_q = B * NQ;                        // 829440
    float* ccost = (float*)d_ws;                       // total_q floats = 3.3 MB

    class_cost_kernel<<<dim3((total_q + 255) / 256), dim3(256), 0, stream>>>(
        pred_logits, ccost, total_q);

    const int npairs = B * TPB;                        // 10240 waves
    match_kernel<<<dim3((npairs * 32 + 127) / 128), dim3(128), 0, stream>>>(
        pred_boxes, tgt_boxes, tgt_labels, ccost, out, npairs);
}